// ChannelSpatialContextAttention_66769561584100
// MI455X (gfx1250) — compile-verified
//
#include <hip/hip_runtime.h>
#include <hip/hip_bf16.h>

typedef __attribute__((ext_vector_type(2))) float v2f;
typedef __attribute__((ext_vector_type(8))) float v8f;

#define Bc 8
#define DIMc 512
#define Hc 128
#define Wc 128
#define HWc 16384
#define Mc 32
#define EPSc 1e-5f
#define COORD_TAUc 1.0f
#define COORD_KAPPAc 1.0f
#define COT_TAUc 0.8f
#define COT_LAMc 0.7f
#define GATE_FLOORc 0.05f

// ---------------- K1: compress GEMM  D[32,HW] = W[32,512] @ X[512,HW] (per batch) --------
__global__ __launch_bounds__(256) void k_compress(const float* __restrict__ x,
                                                  const float* __restrict__ w,
                                                  float* __restrict__ xcp) {
  __shared__ float ldsw[512 * 32];  // [k][c]
  int t = threadIdx.x;
  for (int i = t; i < 512 * 32; i += 256) {
    int k = i >> 5, c = i & 31;
    ldsw[i] = w[c * 512 + k];
  }
  __syncthreads();
  int wave = t >> 5, lane = t & 31;
  int half = lane >> 4, r = lane & 15;
  int pixBlk = blockIdx.x % (HWc / 128);
  int b = blockIdx.x / (HWc / 128);
  int pbase = pixBlk * 128 + wave * 16;
  const float* X = x + (size_t)b * DIMc * HWc;
  v8f c0 = {}; v8f c1 = {};
  for (int k0 = 0; k0 < 512; k0 += 4) {
    int kk = k0 + 2 * half;
    v2f bm;
    bm.x = X[(size_t)kk * HWc + pbase + r];
    bm.y = X[(size_t)(kk + 1) * HWc + pbase + r];
    v2f a0, a1;
    a0.x = ldsw[kk * 32 + r];       a0.y = ldsw[(kk + 1) * 32 + r];
    a1.x = ldsw[kk * 32 + 16 + r];  a1.y = ldsw[(kk + 1) * 32 + 16 + r];
    c0 = __builtin_amdgcn_wmma_f32_16x16x4_f32(false, a0, false, bm, (short)0, c0, false, false);
    c1 = __builtin_amdgcn_wmma_f32_16x16x4_f32(false, a1, false, bm, (short)0, c1, false, false);
  }
  float* O = xcp + (size_t)b * Mc * HWc;
#pragma unroll
  for (int vI = 0; vI < 8; vI++) {
    int mrow = vI + 8 * half;
    O[(size_t)mrow * HWc + pbase + r] = c0[vI];
    O[(size_t)(16 + mrow) * HWc + pbase + r] = c1[vI];
  }
}

// ---------------- K2: per-(b,c) instance-norm stats -> fused scale/shift -----------------
__global__ __launch_bounds__(256) void k_stats(const float* __restrict__ src,
                                               const float* __restrict__ gamma,
                                               const float* __restrict__ beta,
                                               float* __restrict__ sc, float* __restrict__ sh) {
  int row = blockIdx.x;  // b*32 + c
  int c = row & 31;
  const float* p = src + (size_t)row * HWc;
  float s = 0.f, s2 = 0.f;
  for (int i = threadIdx.x; i < HWc; i += 256) { float v = p[i]; s += v; s2 += v * v; }
  __shared__ float rs[256], rs2[256];
  rs[threadIdx.x] = s; rs2[threadIdx.x] = s2;
  __syncthreads();
  for (int off = 128; off > 0; off >>= 1) {
    if (threadIdx.x < off) { rs[threadIdx.x] += rs[threadIdx.x + off]; rs2[threadIdx.x] += rs2[threadIdx.x + off]; }
    __syncthreads();
  }
  if (threadIdx.x == 0) {
    float mu = rs[0] * (1.f / HWc);
    float var = rs2[0] * (1.f / HWc) - mu * mu;
    float rsig = rsqrtf(var + EPSc);
    float g = gamma[c];
    sc[row] = rsig * g;
    sh[row] = beta[c] - mu * rsig * g;
  }
}

// ---------------- K3: pooled H/W means of xc (norm applied inline) -----------------------
__global__ __launch_bounds__(128) void k_pool(const float* __restrict__ xcp,
                                              const float* __restrict__ sc,
                                              const float* __restrict__ sh,
                                              float* __restrict__ xh, float* __restrict__ xw) {
  int row = blockIdx.x;
  const float* p = xcp + (size_t)row * HWc;
  float scv = sc[row], shv = sh[row];
  int t = threadIdx.x;
  float s = 0.f;
  for (int wI = 0; wI < Wc; wI++) s += fmaxf(fmaf(p[t * Wc + wI], scv, shv), 0.f);
  xh[(size_t)row * Hc + t] = s * (1.f / Wc);
  float s2 = 0.f;
  for (int hI = 0; hI < Hc; hI++) s2 += fmaxf(fmaf(p[hI * Wc + t], scv, shv), 0.f);
  xw[(size_t)row * Wc + t] = s2 * (1.f / Hc);
}

// ---------------- K4: coord branch (both axes), one block per batch ----------------------
__device__ __forceinline__ void coord_branch(int b, int t, const float* pooled,
                                             const float* pw, const float* pg, const float* pb,
                                             const float* zw, const float* zb, float* outA,
                                             float* bufT, float* bufU, float* chS, float* chT) {
  const int L = 128;
  for (int i = t; i < 32 * L; i += 128) bufT[i] = pooled[(size_t)b * 32 * L + i];
  __syncthreads();
  for (int i = t; i < 32 * L; i += 128) {
    int cI = i / L, pI = i % L;
    int g = cI >> 3, o = cI & 7;
    float acc = 0.f;
    for (int j = 0; j < 8; j++) acc = fmaf(pw[(g * 8 + o) * 8 + j], bufT[(g * 8 + j) * L + pI], acc);
    bufU[i] = acc;
  }
  __syncthreads();
  if (t < 32) {
    float s = 0.f, s2 = 0.f;
    for (int pI = 0; pI < L; pI++) { float v = bufU[t * L + pI]; s += v; s2 += v * v; }
    float mu = s * (1.f / L), var = s2 * (1.f / L) - mu * mu;
    float rsig = rsqrtf(var + EPSc);
    chS[t] = rsig * pg[t];
    chT[t] = pb[t] - mu * rsig * pg[t];
  }
  __syncthreads();
  for (int i = t; i < 32 * L; i += 128) {
    int cI = i / L;
    float u = fmaf(bufU[i], chS[cI], chT[cI]);
    bufT[i] = u / (1.f + __expf(-u));  // silu
  }
  __syncthreads();
  for (int i = t; i < 32 * L; i += 128) {
    int cI = i / L, pI = i % L;
    int g = cI >> 3, o = cI & 7;
    float acc = zb[cI];
    for (int j = 0; j < 8; j++) acc = fmaf(zw[(g * 8 + o) * 8 + j], bufT[(g * 8 + j) * L + pI], acc);
    bufU[i] = acc * (1.f / COORD_TAUc);
  }
  __syncthreads();
  if (t < 32) {
    float mx = -1e30f;
    for (int pI = 0; pI < L; pI++) mx = fmaxf(mx, bufU[t * L + pI]);
    float sum = 0.f;
    for (int pI = 0; pI < L; pI++) sum += __expf(bufU[t * L + pI] - mx);
    float scl = (float)L / sum;
    for (int pI = 0; pI < L; pI++)
      outA[((size_t)b * 32 + t) * L + pI] = __expf(bufU[t * L + pI] - mx) * scl;
  }
  __syncthreads();
}

__global__ __launch_bounds__(128) void k_coord(const float* __restrict__ xh, const float* __restrict__ xw,
                                               const float* __restrict__ pw, const float* __restrict__ pg,
                                               const float* __restrict__ pb,
                                               const float* __restrict__ hwt, const float* __restrict__ hbt,
                                               const float* __restrict__ wwt, const float* __restrict__ wbt,
                                               float* __restrict__ ah, float* __restrict__ aw) {
  __shared__ float bufT[32 * 128], bufU[32 * 128], chS[32], chT[32];
  int b = blockIdx.x, t = threadIdx.x;
  coord_branch(b, t, xh, pw, pg, pb, hwt, hbt, ah, bufT, bufU, chS, chT);
  coord_branch(b, t, xw, pw, pg, pb, wwt, wbt, aw, bufT, bufU, chS, chT);
}

// ---------------- K5: CoT part 1: k1, v, att_pre (per pixel) -----------------------------
__global__ __launch_bounds__(256) void k_cot1(const float* __restrict__ xcp,
                                              const float* __restrict__ sc, const float* __restrict__ sh,
                                              const float* __restrict__ keyw, const float* __restrict__ valw,
                                              const float* __restrict__ a1w,
                                              float* __restrict__ k1o, float* __restrict__ vo,
                                              float* __restrict__ ao) {
  __shared__ float kw[256], vw[256], aw1[512];
  int t = threadIdx.x;
  kw[t] = keyw[t]; vw[t] = valw[t];
  for (int i = t; i < 512; i += 256) aw1[i] = a1w[i];
  __syncthreads();
  size_t pix = (size_t)blockIdx.x * 256 + t;
  int b = (int)(pix / HWc), p = (int)(pix % HWc);
  size_t off = (size_t)b * Mc * HWc + p;
  float xv[32];
#pragma unroll
  for (int cI = 0; cI < 32; cI++)
    xv[cI] = fmaxf(fmaf(xcp[off + (size_t)cI * HWc], sc[b * 32 + cI], sh[b * 32 + cI]), 0.f);
  float k1[32], vv[32];
#pragma unroll
  for (int g = 0; g < 4; g++)
#pragma unroll
    for (int o = 0; o < 8; o++) {
      float ak = 0.f, av = 0.f;
#pragma unroll
      for (int j = 0; j < 8; j++) {
        float xj = xv[g * 8 + j];
        ak = fmaf(kw[(g * 8 + o) * 8 + j], xj, ak);
        av = fmaf(vw[(g * 8 + o) * 8 + j], xj, av);
      }
      k1[g * 8 + o] = ak; vv[g * 8 + o] = av;
    }
  // att1 on concat [k1(32); xc(32)] viewed as 4 groups of 16
  float at[32];
#pragma unroll
  for (int g = 0; g < 4; g++)
#pragma unroll
    for (int o = 0; o < 8; o++) {
      float acc = 0.f;
#pragma unroll
      for (int j = 0; j < 16; j++) {
        int ch = g * 16 + j;
        float in = (ch < 32) ? k1[ch] : xv[ch - 32];
        acc = fmaf(aw1[(g * 8 + o) * 16 + j], in, acc);
      }
      at[g * 8 + o] = acc;
    }
#pragma unroll
  for (int cI = 0; cI < 32; cI++) {
    k1o[off + (size_t)cI * HWc] = k1[cI];
    vo[off + (size_t)cI * HWc] = vv[cI];
    ao[off + (size_t)cI * HWc] = at[cI];
  }
}

// ---------------- K6: attn-norm + att2 gconv (in place on att buffer) --------------------
__global__ __launch_bounds__(256) void k_cot2(float* att,
                                              const float* __restrict__ sc2, const float* __restrict__ sh2,
                                              const float* __restrict__ a2w) {
  __shared__ float w2[256];
  int t = threadIdx.x;
  w2[t] = a2w[t];
  __syncthreads();
  size_t pix = (size_t)blockIdx.x * 256 + t;
  int b = (int)(pix / HWc), p = (int)(pix % HWc);
  size_t off = (size_t)b * Mc * HWc + p;
  float at[32];
#pragma unroll
  for (int cI = 0; cI < 32; cI++)
    at[cI] = fmaxf(fmaf(att[off + (size_t)cI * HWc], sc2[b * 32 + cI], sh2[b * 32 + cI]), 0.f);
#pragma unroll
  for (int g = 0; g < 4; g++)
#pragma unroll
    for (int o = 0; o < 8; o++) {
      float acc = 0.f;
#pragma unroll
      for (int j = 0; j < 8; j++) acc = fmaf(w2[(g * 8 + o) * 8 + j], at[g * 8 + j], acc);
      att[off + (size_t)(g * 8 + o) * HWc] = acc;
    }
}

// ---------------- K7: spatial softmax stats per (b,c) ------------------------------------
__global__ __launch_bounds__(256) void k_smstats(const float* __restrict__ lg,
                                                 float* __restrict__ rmax, float* __restrict__ rsum) {
  int row = blockIdx.x;
  const float* p = lg + (size_t)row * HWc;
  __shared__ float red[256];
  float m = -1e30f;
  for (int i = threadIdx.x; i < HWc; i += 256) m = fmaxf(m, p[i] * (1.f / COT_TAUc));
  red[threadIdx.x] = m; __syncthreads();
  for (int off = 128; off > 0; off >>= 1) {
    if (threadIdx.x < off) red[threadIdx.x] = fmaxf(red[threadIdx.x], red[threadIdx.x + off]);
    __syncthreads();
  }
  float mx = red[0];
  __syncthreads();
  float s = 0.f;
  for (int i = threadIdx.x; i < HWc; i += 256) s += __expf(p[i] * (1.f / COT_TAUc) - mx);
  red[threadIdx.x] = s; __syncthreads();
  for (int off = 128; off > 0; off >>= 1) {
    if (threadIdx.x < off) red[threadIdx.x] += red[threadIdx.x + off];
    __syncthreads();
  }
  if (threadIdx.x == 0) { rmax[row] = mx; rsum[row] = red[0]; }
}

// ---------------- K8: att_mix + cot-fuse + coord_out + fusion (per pixel) ----------------
__global__ __launch_bounds__(256) void k_mix(const float* __restrict__ xcp,
                                             const float* __restrict__ sc, const float* __restrict__ sh,
                                             const float* __restrict__ k1i, const float* __restrict__ vi,
                                             const float* __restrict__ lgi,
                                             const float* __restrict__ rmax, const float* __restrict__ rsum,
                                             const float* __restrict__ ah, const float* __restrict__ aw,
                                             const float* __restrict__ alpha, const float* __restrict__ beta,
                                             const float* __restrict__ fusew, const float* __restrict__ fusionw,
                                             float* __restrict__ fpre) {
  __shared__ float fw[1024], uw[2048];
  int t = threadIdx.x;
  for (int i = t; i < 1024; i += 256) fw[i] = fusew[i];
  for (int i = t; i < 2048; i += 256) uw[i] = fusionw[i];
  __syncthreads();
  size_t pix = (size_t)blockIdx.x * 256 + t;
  int b = (int)(pix / HWc), p = (int)(pix % HWc);
  int hI = p / Wc, wI = p % Wc;
  float al = alpha[0], be = beta[0];
  size_t off = (size_t)b * Mc * HWc + p;
  float cin[32], co[32];
#pragma unroll
  for (int cI = 0; cI < 32; cI++) {
    int row = b * 32 + cI;
    float lg = lgi[off + (size_t)cI * HWc];
    float z = lg * (1.f / COT_TAUc);
    float sm = __expf(z - rmax[row]) / rsum[row] * (float)HWc;
    float sig = 1.f / (1.f + __expf(-lg));
    float amix = (1.f - COT_LAMc) * sig + COT_LAMc * sm;
    cin[cI] = fmaf(amix, vi[off + (size_t)cI * HWc], k1i[off + (size_t)cI * HWc]);
    float xcv = fmaxf(fmaf(xcp[off + (size_t)cI * HWc], sc[row], sh[row]), 0.f);
    float wlin = al * ah[(size_t)row * Hc + hI] + be * aw[(size_t)row * Wc + wI];
    co[cI] = xcv * (1.f + COORD_KAPPAc * (wlin - 1.f));
  }
  float cot[32];
#pragma unroll
  for (int o = 0; o < 32; o++) {
    float acc = 0.f;
#pragma unroll
    for (int i = 0; i < 32; i++) acc = fmaf(fw[o * 32 + i], cin[i], acc);
    cot[o] = acc;
  }
#pragma unroll
  for (int o = 0; o < 32; o++) {
    float acc = 0.f;
#pragma unroll
    for (int i = 0; i < 32; i++) acc = fmaf(uw[o * 64 + i], co[i], acc);
#pragma unroll
    for (int i = 0; i < 32; i++) acc = fmaf(uw[o * 64 + 32 + i], cot[i], acc);
    fpre[off + (size_t)o * HWc] = acc;
  }
}

// ---------------- K9: apply fusion norm + relu (in place) --------------------------------
__global__ __launch_bounds__(256) void k_apply(float* buf, const float* __restrict__ sc3,
                                               const float* __restrict__ sh3) {
  size_t i = (size_t)blockIdx.x * 256 + threadIdx.x;
  int row = (int)(i / HWc);
  buf[i] = fmaxf(fmaf(buf[i], sc3[row], sh3[row]), 0.f);
}

// ---------------- K10: per-(b,c) spatial mean of fused -----------------------------------
__global__ __launch_bounds__(256) void k_rowmean(const float* __restrict__ src, float* __restrict__ mean) {
  int row = blockIdx.x;
  const float* p = src + (size_t)row * HWc;
  __shared__ float red[256];
  float s = 0.f;
  for (int i = threadIdx.x; i < HWc; i += 256) s += p[i];
  red[threadIdx.x] = s; __syncthreads();
  for (int off = 128; off > 0; off >>= 1) {
    if (threadIdx.x < off) red[threadIdx.x] += red[threadIdx.x + off];
    __syncthreads();
  }
  if (threadIdx.x == 0) mean[row] = red[0] * (1.f / HWc);
}

// ---------------- K11: gate scalars per batch --------------------------------------------
__global__ void k_gate(const float* __restrict__ fmean, const float* __restrict__ gw,
                       float* __restrict__ gs) {
  int b = threadIdx.x;
  if (b < Bc) {
    float acc = 0.f;
    for (int cI = 0; cI < 32; cI++) acc = fmaf(gw[cI], fmean[b * 32 + cI], acc);
    float g = 1.f / (1.f + __expf(-acc));
    gs[b] = g * (1.f - GATE_FLOORc) + GATE_FLOORc;
  }
}

// ---------------- K12: expand GEMM  out[512,HW] = W[512,32] @ (gate*fused)[32,HW] --------
__global__ __launch_bounds__(256) void k_expand(const float* __restrict__ fused,
                                                const float* __restrict__ ew,
                                                const float* __restrict__ gs,
                                                float* __restrict__ out) {
  __shared__ float ldsw[32 * 512];  // [k][o]
  int t = threadIdx.x;
  for (int i = t; i < 32 * 512; i += 256) {
    int o = i & 511, k = i >> 9;
    ldsw[i] = ew[o * 32 + k];
  }
  __syncthreads();
  int wave = t >> 5, lane = t & 31;
  int half = lane >> 4, r = lane & 15;
  int nb = blockIdx.x % (HWc / 32);
  int b = blockIdx.x / (HWc / 32);
  int pbase = nb * 32;
  float g = gs[b];
  const float* F = fused + (size_t)b * Mc * HWc;
  v8f acc[4][2] = {};
#pragma unroll
  for (int k0 = 0; k0 < 32; k0 += 4) {
    int kk = k0 + 2 * half;
    v2f b0, b1;
    b0.x = F[(size_t)kk * HWc + pbase + r] * g;
    b0.y = F[(size_t)(kk + 1) * HWc + pbase + r] * g;
    b1.x = F[(size_t)kk * HWc + pbase + 16 + r] * g;
    b1.y = F[(size_t)(kk + 1) * HWc + pbase + 16 + r] * g;
#pragma unroll
    for (int mt = 0; mt < 4; mt++) {
      int mbase = wave * 64 + mt * 16;
      v2f a;
      a.x = ldsw[kk * 512 + mbase + r];
      a.y = ldsw[(kk + 1) * 512 + mbase + r];
      acc[mt][0] = __builtin_amdgcn_wmma_f32_16x16x4_f32(false, a, false, b0, (short)0, acc[mt][0], false, false);
      acc[mt][1] = __builtin_amdgcn_wmma_f32_16x16x4_f32(false, a, false, b1, (short)0, acc[mt][1], false, false);
    }
  }
  float* O = out + (size_t)b * DIMc * HWc;
#pragma unroll
  for (int mt = 0; mt < 4; mt++)
#pragma unroll
    for (int vI = 0; vI < 8; vI++) {
      int mrow = wave * 64 + mt * 16 + vI + 8 * half;
      O[(size_t)mrow * HWc + pbase + r] = acc[mt][0][vI];
      O[(size_t)mrow * HWc + pbase + 16 + r] = acc[mt][1][vI];
    }
}

extern "C" void kernel_launch(void* const* d_in, const int* in_sizes, int n_in,
                              void* d_out, int out_size, void* d_ws, size_t ws_size,
                              hipStream_t stream) {
  (void)in_sizes; (void)n_in; (void)out_size; (void)ws_size;
  const float* x            = (const float*)d_in[0];
  const float* compress_w   = (const float*)d_in[1];
  const float* inc_g        = (const float*)d_in[2];
  const float* inc_b        = (const float*)d_in[3];
  const float* coord_proj_w = (const float*)d_in[4];
  const float* coord_proj_g = (const float*)d_in[5];
  const float* coord_proj_b = (const float*)d_in[6];
  const float* coord_h_w    = (const float*)d_in[7];
  const float* coord_h_b    = (const float*)d_in[8];
  const float* coord_wc_w   = (const float*)d_in[9];
  const float* coord_wc_b   = (const float*)d_in[10];
  const float* coord_alpha  = (const float*)d_in[11];
  const float* coord_beta   = (const float*)d_in[12];
  const float* cot_key_w    = (const float*)d_in[13];
  const float* cot_val_w    = (const float*)d_in[14];
  const float* cot_att1_w   = (const float*)d_in[15];
  const float* cot_attn_g   = (const float*)d_in[16];
  const float* cot_attn_b   = (const float*)d_in[17];
  const float* cot_att2_w   = (const float*)d_in[18];
  const float* cot_fuse_w   = (const float*)d_in[19];
  const float* fusion_w     = (const float*)d_in[20];
  const float* fus_g        = (const float*)d_in[21];
  const float* fus_b        = (const float*)d_in[22];
  const float* gate_w       = (const float*)d_in[23];
  const float* expand_w     = (const float*)d_in[24];
  float* out = (float*)d_out;

  float* ws = (float*)d_ws;
  const size_t NBC = (size_t)Bc * Mc * HWc;  // 4,194,304 floats
  float* xcp = ws;
  float* k1b = ws + NBC;
  float* vb  = ws + 2 * NBC;
  float* ab  = ws + 3 * NBC;
  float* fp  = ws + 4 * NBC;
  float* xh  = ws + 5 * NBC;
  float* xw  = xh + (size_t)Bc * Mc * Hc;
  float* ah  = xw + (size_t)Bc * Mc * Wc;
  float* aw  = ah + (size_t)Bc * Mc * Hc;
  float* sc1 = aw + (size_t)Bc * Mc * Wc;
  float* sh1 = sc1 + 256;
  float* sc2 = sh1 + 256;  float* sh2 = sc2 + 256;
  float* sc3 = sh2 + 256;  float* sh3 = sc3 + 256;
  float* rmax = sh3 + 256; float* rsum = rmax + 256;
  float* fmean = rsum + 256; float* gs = fmean + 256;

  k_compress<<<Bc * (HWc / 128), 256, 0, stream>>>(x, compress_w, xcp);
  k_stats<<<Bc * Mc, 256, 0, stream>>>(xcp, inc_g, inc_b, sc1, sh1);
  k_pool<<<Bc * Mc, 128, 0, stream>>>(xcp, sc1, sh1, xh, xw);
  k_coord<<<Bc, 128, 0, stream>>>(xh, xw, coord_proj_w, coord_proj_g, coord_proj_b,
                                  coord_h_w, coord_h_b, coord_wc_w, coord_wc_b, ah, aw);
  k_cot1<<<(Bc * HWc) / 256, 256, 0, stream>>>(xcp, sc1, sh1, cot_key_w, cot_val_w, cot_att1_w,
                                               k1b, vb, ab);
  k_stats<<<Bc * Mc, 256, 0, stream>>>(ab, cot_attn_g, cot_attn_b, sc2, sh2);
  k_cot2<<<(Bc * HWc) / 256, 256, 0, stream>>>(ab, sc2, sh2, cot_att2_w);
  k_smstats<<<Bc * Mc, 256, 0, stream>>>(ab, rmax, rsum);
  k_mix<<<(Bc * HWc) / 256, 256, 0, stream>>>(xcp, sc1, sh1, k1b, vb, ab, rmax, rsum,
                                              ah, aw, coord_alpha, coord_beta,
                                              cot_fuse_w, fusion_w, fp);
  k_stats<<<Bc * Mc, 256, 0, stream>>>(fp, fus_g, fus_b, sc3, sh3);
  k_apply<<<(int)((Bc * (size_t)Mc * HWc) / 256), 256, 0, stream>>>(fp, sc3, sh3);
  k_rowmean<<<Bc * Mc, 256, 0, stream>>>(fp, fmean);
  k_gate<<<1, 32, 0, stream>>>(fmean, gate_w, gs);
  k_expand<<<Bc * (HWc / 32), 256, 0, stream>>>(fp, expand_w, gs, out);
}